// InteractionArch_82497731821575
// MI455X (gfx1250) — compile-verified
//
#include <hip/hip_runtime.h>

// CDNA5 / gfx1250, wave32.
typedef float v2f __attribute__((ext_vector_type(2)));
typedef float v4f __attribute__((ext_vector_type(4)));
typedef float v8f __attribute__((ext_vector_type(8)));

#define F_FEAT 26
#define D_DIM 128
#define OUT_STRIDE 479   // 128 dense + 26 sd + 325 pairs
#define SD_BASE 128
#define PACK_N 351       // 26 sd + 325 pairs (out[128..478])
#define WAVES_PER_BLOCK 8

// Scatter one Gram entry G[i][j] (virtual 27x27, row 0 = dense) into the
// per-wave packed staging buffer: stage[0..25] = sd, stage[26..350] = pairs.
__device__ __forceinline__ void stage_gram(float* __restrict__ stage, int i, int j, float val) {
    if (j > F_FEAT) return;                 // padded column
    if (i == 0) {                           // sd = sparse . dense
        if (j >= 1) stage[j - 1] = val;
        return;
    }
    if (i < j) {                            // strict upper triangle, row-major (triu k=1)
        const int a = i - 1;                // 0..24
        const int c = j - 1;                // a+1..25
        stage[F_FEAT + (a * (2 * F_FEAT - a - 1)) / 2 + (c - a - 1)] = val;
    }
}

__global__ __launch_bounds__(WAVES_PER_BLOCK * 32) void dlrm_interaction_wmma(
    const float* __restrict__ dense,    // [B,128]
    const float* __restrict__ sparse,   // [B,26,128]
    float* __restrict__ out,            // [B,479]
    int B)
{
    __shared__ float smem[WAVES_PER_BLOCK][PACK_N + 1];   // +1 pad vs bank aliasing

    const int lane = threadIdx.x & 31;
    const int w    = threadIdx.x >> 5;
    const int b    = blockIdx.x * WAVES_PER_BLOCK + w;
    if (b >= B) return;                 // whole-wave uniform exit

    const int r  = lane & 15;           // fragment row/col 0..15
    const int h  = lane >> 4;           // half-wave selects K sub-pair
    const int ko = 2 * h;

    // Virtual matrix T: row 0 = dense[b], rows 1..26 = sparse[b][row-1], rows 27..31 = 0.
    const float* rowLo = (r == 0)
        ? dense + (size_t)b * D_DIM
        : sparse + ((size_t)b * F_FEAT + (r - 1)) * D_DIM;
    const bool  hiValid = (r + 16) <= F_FEAT;                    // T rows 16..26
    const float hiMask  = hiValid ? 1.0f : 0.0f;                 // branchless pad (EXEC stays all-1s)
    const float* rowHi  = sparse + ((size_t)b * F_FEAT + (hiValid ? (r + 15) : 0)) * D_DIM;

    v8f c00 = {}; v8f c01 = {}; v8f c11 = {};

    // ISA fp32 A 16x4 fragment: lane l -> row l&15, VGPR0/1 = K kb+2h, kb+2h+1.
    // B 4x16 fragment (C/D striping rule) has the identical per-lane layout for
    // B = A^T, so one float2 load serves as both operands.
    #pragma unroll 4
    for (int k = 0; k < D_DIM; k += 4) {
        v2f aLo = *(const v2f*)(rowLo + k + ko);
        v2f t   = *(const v2f*)(rowHi + k + ko);
        v2f aHi; aHi.x = t.x * hiMask; aHi.y = t.y * hiMask;

        c00 = __builtin_amdgcn_wmma_f32_16x16x4_f32(false, aLo, false, aLo, (short)0, c00, false, false);
        c01 = __builtin_amdgcn_wmma_f32_16x16x4_f32(false, aLo, false, aHi, (short)0, c01, false, false);
        c11 = __builtin_amdgcn_wmma_f32_16x16x4_f32(false, aHi, false, aHi, (short)0, c11, false, false);
    }

    float* orow  = out + (size_t)b * OUT_STRIDE;
    float* stage = smem[w];

    // Dense passthrough: coalesced float4 copy (32 lanes x 16B = 512B).
    *(v4f*)(orow + 4 * lane) = *(const v4f*)(dense + (size_t)b * D_DIM + 4 * lane);

    // Scatter accumulators into LDS staging (predicated ds stores, same-wave only).
    // C/D layout: VGPR v, lane l -> G_tile[v + 8*(l>>4)][l&15].
    #pragma unroll
    for (int v = 0; v < 8; ++v) {
        const int iLo = v + 8 * h;      // 0..15
        const int iHi = iLo + 16;       // 16..31
        const int jLo = r;              // 0..15
        const int jHi = r + 16;         // 16..31
        stage_gram(stage, iLo, jLo, c00[v]);
        stage_gram(stage, iLo, jHi, c01[v]);
        stage_gram(stage, iHi, jHi, c11[v]);
    }

    // Same-wave LDS RAW: wait for all ds stores to land before re-reading.
    asm volatile("s_wait_dscnt 0x0" ::: "memory");

    // Packed, fully-coalesced writeout of out[b][128..478] (351 floats).
    #pragma unroll
    for (int t = 0; t < 11; ++t) {
        const int idx = t * 32 + lane;
        if (idx < PACK_N) orow[SD_BASE + idx] = stage[idx];
    }
}

extern "C" void kernel_launch(void* const* d_in, const int* in_sizes, int n_in,
                              void* d_out, int out_size, void* d_ws, size_t ws_size,
                              hipStream_t stream) {
    const float* dense  = (const float*)d_in[0];   // [B,128] f32
    const float* sparse = (const float*)d_in[1];   // [B,26,128] f32
    float* out = (float*)d_out;                    // [B,479] f32

    const int B = in_sizes[0] / D_DIM;
    const int threads = WAVES_PER_BLOCK * 32;      // 8 wave32 waves -> 8 samples/block
    const int blocks = (B + WAVES_PER_BLOCK - 1) / WAVES_PER_BLOCK;

    dlrm_interaction_wmma<<<blocks, threads, 0, stream>>>(dense, sparse, out, B);
}